// AttentionPooling_33809982554590
// MI455X (gfx1250) — compile-verified
//
#include <hip/hip_runtime.h>
#include <hip/hip_bf16.h>
#include <math.h>

#define D      128
#define SEQ    512
#define NB     2048
#define SCALE  0.08838834764831845f   // 128^-0.5

typedef __attribute__((ext_vector_type(2))) float v2f;
typedef __attribute__((ext_vector_type(8))) float v8f;

// ---------------------------------------------------------------------------
// Kernel 1: fold query through the K-path.
//   qk[d] = sum_k q[k]*Wk[k,d]   (ws[0..127])
//   qp[p] = sum_k q[k]*Wp[k,p]   (ws[128..131])
//   c     = q·(bk+bp)            (ws[132])
// ---------------------------------------------------------------------------
__global__ __launch_bounds__(128) void ap_prep_kernel(
    const float* __restrict__ q,  const float* __restrict__ Wk,
    const float* __restrict__ bk, const float* __restrict__ Wp,
    const float* __restrict__ bp, float* __restrict__ ws) {
  __shared__ float qs[D];
  const int t = threadIdx.x;
  qs[t] = q[t];
  __syncthreads();
  float acc = 0.f;
  for (int k = 0; k < D; ++k) acc += qs[k] * Wk[k * D + t];   // coalesced cols
  ws[t] = acc;
  if (t < 4) {
    float a = 0.f;
    for (int k = 0; k < D; ++k) a += qs[k] * Wp[k * 4 + t];
    ws[D + t] = a;
  }
  if (t == 4) {
    float a = 0.f;
    for (int k = 0; k < D; ++k) a += qs[k] * (bk[k] + bp[k]);
    ws[D + 4] = a;
  }
}

// ---------------------------------------------------------------------------
// Kernel 2: one block per batch row. Streaming, bandwidth-bound.
//   pass1: scores[s] = (x[b,s]·qk + pos[b,s]·qp + c)*SCALE  (masked)
//   softmax over S in LDS
//   pass2: xbar[b,d] = sum_s softmax(s) * x[b,s,d]
// 256 threads = 8 waves (wave32). Each wave owns tokens s ≡ wave (mod 8);
// each lane owns a contiguous float4 of the 128-dim token (fully coalesced
// 512B/wave/token), reduced with 5 __shfl_xor steps.
// ---------------------------------------------------------------------------
__global__ __launch_bounds__(256) void ap_pool_kernel(
    const float* __restrict__ x, const float* __restrict__ pos,
    const unsigned char* __restrict__ mask,
    const float* __restrict__ cons, float* __restrict__ xbar) {
  __shared__ float scores[SEQ];
  __shared__ float partial[8 * D];
  __shared__ float red[256];

  const int b = blockIdx.x;
  const int t = threadIdx.x;
  const int w = t >> 5;     // wave 0..7
  const int l = t & 31;     // lane 0..31

  const float4 qk4 = *(const float4*)&cons[l * 4];
  const float4 qp4 = *(const float4*)&cons[D];
  const float  cc  = cons[D + 4];
  const float* xb  = x + (size_t)b * SEQ * D;

  // ---- pass 1: scores ----
  for (int s = w; s < SEQ; s += 8) {
    const float4 x4 = *(const float4*)&xb[s * D + l * 4];
    float p = x4.x * qk4.x + x4.y * qk4.y + x4.z * qk4.z + x4.w * qk4.w;
#pragma unroll
    for (int off = 16; off >= 1; off >>= 1) p += __shfl_xor(p, off, 32);
    if (l == 0) {
      const float4 p4 = *(const float4*)&pos[((size_t)b * SEQ + s) * 4];
      float sc = (p + p4.x * qp4.x + p4.y * qp4.y + p4.z * qp4.z +
                  p4.w * qp4.w + cc) * SCALE;
      if (!mask[(size_t)b * SEQ + s]) sc = -INFINITY;
      scores[s] = sc;
    }
  }
  __syncthreads();

  // ---- softmax: max ----
  red[t] = fmaxf(scores[t], scores[t + 256]);
  __syncthreads();
  for (int st = 128; st >= 1; st >>= 1) {
    if (t < st) red[t] = fmaxf(red[t], red[t + st]);
    __syncthreads();
  }
  const float smax = red[0];
  __syncthreads();

  // ---- softmax: exp + sum ----
  const float e0 = __expf(scores[t] - smax);
  const float e1 = __expf(scores[t + 256] - smax);
  scores[t] = e0;
  scores[t + 256] = e1;
  red[t] = e0 + e1;
  __syncthreads();
  for (int st = 128; st >= 1; st >>= 1) {
    if (t < st) red[t] += red[t + st];
    __syncthreads();
  }
  const float inv = 1.0f / red[0];

  // ---- pass 2: weighted sum over tokens (x re-read hits L2) ----
  float4 acc = make_float4(0.f, 0.f, 0.f, 0.f);
  for (int s = w; s < SEQ; s += 8) {
    const float4 x4 = *(const float4*)&xb[s * D + l * 4];
    const float wgt = scores[s];
    acc.x += wgt * x4.x; acc.y += wgt * x4.y;
    acc.z += wgt * x4.z; acc.w += wgt * x4.w;
  }
  *(float4*)&partial[w * D + l * 4] = acc;
  __syncthreads();
  if (t < D) {
    float s = 0.f;
#pragma unroll
    for (int ww = 0; ww < 8; ++ww) s += partial[ww * D + t];
    xbar[(size_t)b * D + t] = s * inv;
  }
}

// ---------------------------------------------------------------------------
// Kernel 3: pooled = xbar · Wv^T + bv  via V_WMMA_F32_16X16X4_F32.
// M=2048 (batches), N=128 (out dim), K=128. 128 blocks, 8 waves/block;
// wave `wv` computes the 16x16 tile (m0 = 16*blockIdx, n0 = 16*wv),
// 32 WMMA steps over K.
// ISA layout (32-bit 16x4 A / 4x16 B): lanes 0-15 hold K={kb,kb+1},
// lanes 16-31 hold K={kb+2,kb+3} -> one contiguous float2 per lane/step.
// C/D: vgpr r: lanes0-15 -> row r, lanes16-31 -> row r+8, col = lane%16.
// ---------------------------------------------------------------------------
__global__ __launch_bounds__(256) void ap_gemm_kernel(
    const float* __restrict__ xbar, const float* __restrict__ Wv,
    const float* __restrict__ bvec, float* __restrict__ out) {
  const int t    = threadIdx.x;
  const int wv   = t >> 5;
  const int l    = t & 31;
  const int half = l >> 4;     // 0: K lo pair, 1: K hi pair
  const int lm   = l & 15;
  const int m0   = blockIdx.x * 16;
  const int n0   = wv * 16;

  const float* arow = xbar + (size_t)(m0 + lm) * D;  // A[m][k] = xbar[m,k]
  const float* brow = Wv   + (size_t)(n0 + lm) * D;  // B[k][n] = Wv[n,k]
  const int koff = half * 2;

  v8f acc = {0.f, 0.f, 0.f, 0.f, 0.f, 0.f, 0.f, 0.f};
#pragma unroll 4
  for (int kb = 0; kb < D; kb += 4) {
    const v2f a = *(const v2f*)&arow[kb + koff];
    const v2f bb = *(const v2f*)&brow[kb + koff];
    // (neg_a, A, neg_b, B, c_mod, C, reuse_a, reuse_b)
    acc = __builtin_amdgcn_wmma_f32_16x16x4_f32(false, a, false, bb,
                                                (short)0, acc, false, false);
  }

  const float bias = bvec[n0 + lm];
#pragma unroll
  for (int r = 0; r < 8; ++r) {
    out[(size_t)(m0 + half * 8 + r) * D + (n0 + lm)] = acc[r] + bias;
  }
}

// ---------------------------------------------------------------------------
extern "C" void kernel_launch(void* const* d_in, const int* in_sizes, int n_in,
                              void* d_out, int out_size, void* d_ws, size_t ws_size,
                              hipStream_t stream) {
  const float* x    = (const float*)d_in[0];          // [B,S,D]
  const float* pos  = (const float*)d_in[1];          // [B,S,4]
  const unsigned char* mask = (const unsigned char*)d_in[2];  // [B,S] bool
  const float* q    = (const float*)d_in[3];          // [1,D]
  const float* Wk   = (const float*)d_in[4];          // [D,D]
  const float* bk   = (const float*)d_in[5];          // [D]
  const float* Wv   = (const float*)d_in[6];          // [D,D]
  const float* bv   = (const float*)d_in[7];          // [D]
  const float* Wp   = (const float*)d_in[8];          // [D,4]
  const float* bp   = (const float*)d_in[9];          // [D]

  float* ws   = (float*)d_ws;      // [0..127]=qk [128..131]=qp [132]=c
  float* xbar = ws + 256;          // [B,D] = 1 MiB
  float* out  = (float*)d_out;     // [B,D]

  ap_prep_kernel<<<1, 128, 0, stream>>>(q, Wk, bk, Wp, bp, ws);
  ap_pool_kernel<<<NB, 256, 0, stream>>>(x, pos, mask, ws, xbar);
  ap_gemm_kernel<<<NB / 16, 256, 0, stream>>>(xbar, Wv, bv, out);
}